// LRMloss_v2_66039417143335
// MI455X (gfx1250) — compile-verified
//
#include <hip/hip_runtime.h>
#include <math.h>

// ---- problem constants (match reference) ----
#define BB   4
#define HH   200
#define WW   176
#define NF   5
#define AP   2
#define FAa  10          // NF*AP
#define FA7  70          // FAa*7
#define HW   (HH*WW)     // 35200
#define NPF  (HW*AP)     // 70400 elements per (batch,frame)
#define NTOT (BB*HH*WW*FAa)   // 1,408,000
#define EPSF 1e-6f

typedef __attribute__((ext_vector_type(2))) float v2f;
typedef __attribute__((ext_vector_type(8))) float v8f;

// ---------------------------------------------------------------------------
// Wave-level sum via V_WMMA_F32_16X16X4_F32.
// A (16x4 f32, 2 VGPRs over 32 lanes) = {lane_partial, 0}; B = ones.
// D[m][n] = sum_k A[m][k] -> row sums.  Lane n (0-15) holds rows 0..7 in its
// 8 D VGPRs, lane n+16 holds rows 8..15; one shfl_xor(16) completes the sum.
// Requires EXEC == all ones at the call site (guaranteed by construction).
// ---------------------------------------------------------------------------
__device__ __forceinline__ float wave_sum32(float v) {
#if __has_builtin(__builtin_amdgcn_wmma_f32_16x16x4_f32)
  v2f a; a[0] = v;    a[1] = 0.0f;
  v2f b; b[0] = 1.0f; b[1] = 1.0f;
  v8f c = {};
  c = __builtin_amdgcn_wmma_f32_16x16x4_f32(
      /*neg_a=*/false, a, /*neg_b=*/false, b,
      /*c_mod=*/(short)0, c, /*reuse_a=*/false, /*reuse_b=*/false);
  float s = 0.0f;
#pragma unroll
  for (int i = 0; i < 8; ++i) s += c[i];
  s += __shfl_xor(s, 16, 32);
  return s;
#else
#pragma unroll
  for (int off = 16; off > 0; off >>= 1) v += __shfl_xor(v, off, 32);
  return v;
#endif
}

// ---------------------------------------------------------------------------
// Kernel A: fused elementwise pass over [B,H,W,FA].
//  - sigmoid(psm) -> cls_pos partial, neg_log, per-frame loss array
//  - Huber only where pos==1 (skips ~99.5% of rm/targets traffic)
//  - block reductions (WMMA) -> global f32 atomics; per-frame pos counts
// grid = 5500 blocks x 256 threads == NTOT exactly (EXEC stays full).
// ---------------------------------------------------------------------------
__global__ void loss_main_kernel(const float* __restrict__ rm,
                                 const float* __restrict__ psm,
                                 const float* __restrict__ pos1,
                                 const float* __restrict__ neg1,
                                 const float* __restrict__ tgt,
                                 float* __restrict__ acc,      // [0]=hub [1]=pos [2]=clsPos
                                 float* __restrict__ posCnt,   // [20]
                                 float* __restrict__ lossArr)  // [B*NF*NPF]
{
  __shared__ float r_h[8], r_p[8], r_c[8];
  __shared__ float fc[BB * NF];
  const int tid = threadIdx.x;
  if (tid < BB * NF) fc[tid] = 0.0f;
  __syncthreads();

  const int i  = blockIdx.x * blockDim.x + tid;   // < NTOT by construction
  const int a  = i % FAa;
  const int t1 = i / FAa;
  const int w  = t1 % WW;
  const int t2 = t1 / WW;
  const int h  = t2 % HH;
  const int b  = t2 / HH;

  const float pos = pos1[i];
  const float neg = neg1[i];

  const int psmIdx = ((b * FAa + a) * HH + h) * WW + w;   // psm is [B,FA,H,W]
  const float x = psm[psmIdx];
  const float p = 1.0f / (1.0f + expf(-x));

  const float clsPosP = -pos * logf(p + EPSF);
  const float negLog  = logf(1.0f - p + EPSF);
  float loss = -neg * negLog;
  if (!(loss > 0.0f)) loss = 0.0f;   // canonical +0 (kills -0.0 bit pattern for radix select)

  const int f  = a >> 1;
  const int ap = a & 1;
  lossArr[(b * NF + f) * NPF + (h * WW + w) * AP + ap] = loss;

  float hub = 0.0f;
  if (pos > 0.0f) {
    const int rmBase = ((b * FA7 + a * 7) * HH + h) * WW + w;   // rm: [B,FA*7,H,W]
    const int tBase  = ((b * HH + h) * WW + w) * FA7 + a * 7;   // targets: [B,H,W,FA*7]
#pragma unroll
    for (int c = 0; c < 7; ++c) {
      const float d = fabsf(rm[rmBase + c * HW] - tgt[tBase + c]);
      hub += (d < 1.0f) ? 0.5f * d * d : d - 0.5f;
    }
    atomicAdd(&fc[b * NF + f], 1.0f);
  }
  // (branch reconverged: EXEC full again for WMMA reductions)

  const int lane = tid & 31, wv = tid >> 5;
  const float sh = wave_sum32(hub);
  const float sp = wave_sum32(pos);
  const float sc = wave_sum32(clsPosP);
  if (lane == 0) { r_h[wv] = sh; r_p[wv] = sp; r_c[wv] = sc; }
  __syncthreads();
  if (tid == 0) {
    float th = 0.0f, tp = 0.0f, tc = 0.0f;
#pragma unroll
    for (int k = 0; k < 8; ++k) { th += r_h[k]; tp += r_p[k]; tc += r_c[k]; }
    atomicAdd(&acc[0], th);
    atomicAdd(&acc[1], tp);
    atomicAdd(&acc[2], tc);
  }
  if (tid < BB * NF) atomicAdd(&posCnt[tid], fc[tid]);
}

// ---------------------------------------------------------------------------
// Kernel B: per-(batch,frame) hard-negative mining. One block per frame.
// Exact k-th-largest via 4-pass 8-bit radix select on nonneg float bits
// (monotone as uint32), then a final scan: count/sum of loss > T.
// Ties at T>0 have neg==1 and -neg_log == T, so tie contribution = (k-cnt)*T.
// NPF = 275 * 256 exactly -> uniform loop trips, EXEC full at WMMA sites.
// ---------------------------------------------------------------------------
__global__ void mine_kernel(const float* __restrict__ lossArr,
                            const float* __restrict__ posCnt,
                            float* __restrict__ acc)          // [3]=negNum [4]=negDen
{
  __shared__ unsigned int hist[256];
  __shared__ unsigned int sdigit, srem;
  __shared__ float rc[8], rs[8];

  const int bf = blockIdx.x;                     // 0..19
  const float* L = lossArr + (size_t)bf * NPF;

  const float pc = posCnt[bf];
  unsigned int k = (unsigned int)floorf(3.0f * (pc + 1.0f));  // NEG_RATIO*(pos+1)
  if (k > (unsigned int)NPF) k = NPF;
  if (k < 1u) k = 1u;

  unsigned int prefix = 0u, hmask = 0u, remaining = k;
#pragma unroll
  for (int pass = 0; pass < 4; ++pass) {
    const int shift = 24 - 8 * pass;
    for (int bn = threadIdx.x; bn < 256; bn += blockDim.x) hist[bn] = 0u;
    __syncthreads();
    for (int i = threadIdx.x; i < NPF; i += blockDim.x) {
      const unsigned int u = __float_as_uint(L[i]);
      if ((u & hmask) == prefix) atomicAdd(&hist[(u >> shift) & 255u], 1u);
    }
    __syncthreads();
    if (threadIdx.x == 0) {
      unsigned int rem = remaining, dig = 0u;
      for (int bn = 255; bn >= 0; --bn) {
        const unsigned int c = hist[bn];
        if (c >= rem) { dig = (unsigned int)bn; break; }
        rem -= c;
      }
      sdigit = dig; srem = rem;
    }
    __syncthreads();
    prefix |= sdigit << shift;
    hmask  |= 0xFFu << shift;
    remaining = srem;
    __syncthreads();    // before hist reuse next pass
  }
  const float T = __uint_as_float(prefix);       // exact k-th largest value

  // final scan: strictly-greater count and sum (uniform trip count)
  float cnt = 0.0f, sum = 0.0f;
  for (int i = threadIdx.x; i < NPF; i += blockDim.x) {
    const float v = L[i];
    if (v > T) { cnt += 1.0f; sum += v; }
  }
  const int lane = threadIdx.x & 31, wv = threadIdx.x >> 5;
  const float wc = wave_sum32(cnt);
  const float wsm = wave_sum32(sum);
  if (lane == 0) { rc[wv] = wc; rs[wv] = wsm; }
  __syncthreads();
  if (threadIdx.x == 0) {
    float c_gt = 0.0f, s_gt = 0.0f;
#pragma unroll
    for (int j = 0; j < 8; ++j) { c_gt += rc[j]; s_gt += rs[j]; }
    const float num = s_gt + ((float)k - c_gt) * T;   // ties contribute exactly T each
    atomicAdd(&acc[3], num);
    atomicAdd(&acc[4], (float)k);                     // new_neg.sum() = sum of per-frame k
  }
}

// ---------------------------------------------------------------------------
// Kernel C: finalize 4 scalars. d_out = (conf, reg, cls_pos, cls_neg).
// ---------------------------------------------------------------------------
__global__ void finalize_kernel(const float* __restrict__ acc, float* __restrict__ out) {
  if (blockIdx.x == 0 && threadIdx.x == 0) {
    const float hub = acc[0], posSum = acc[1], clsPosSum = acc[2];
    const float negNum = acc[3], negDen = acc[4];
    const float reg = 2.0f * hub / (posSum + EPSF);          // GAMMA
    const float cp  = 1.5f * clsPosSum / (posSum + EPSF);    // ALPHA
    const float cn  = 1.0f * negNum / (negDen + EPSF);       // BETA
    out[0] = cp + cn;   // conf_loss
    out[1] = reg;       // reg_loss
    out[2] = cp;        // cls_pos_loss
    out[3] = cn;        // cls_neg_loss
  }
}

extern "C" void kernel_launch(void* const* d_in, const int* in_sizes, int n_in,
                              void* d_out, int out_size, void* d_ws, size_t ws_size,
                              hipStream_t stream) {
  (void)in_sizes; (void)n_in; (void)out_size; (void)ws_size;
  const float* rm   = (const float*)d_in[0];
  const float* psm  = (const float*)d_in[1];
  const float* pos1 = (const float*)d_in[2];
  const float* neg1 = (const float*)d_in[3];
  const float* tgt  = (const float*)d_in[4];

  float* ws      = (float*)d_ws;
  float* acc     = ws;          // 5 scalars
  float* posCnt  = ws + 8;      // 20 per-frame positive counts
  float* lossArr = ws + 32;     // B*NF*NPF = 1,408,000 floats (~5.6 MB)

  hipMemsetAsync(d_ws, 0, 32 * sizeof(float), stream);   // zero accumulators (capture-safe)

  loss_main_kernel<<<NTOT / 256, 256, 0, stream>>>(rm, psm, pos1, neg1, tgt,
                                                   acc, posCnt, lossArr);
  mine_kernel<<<BB * NF, 256, 0, stream>>>(lossArr, posCnt, acc);
  finalize_kernel<<<1, 64, 0, stream>>>(acc, (float*)d_out);
}